// GNN_GAT_44985487458506
// MI455X (gfx1250) — compile-verified
//
#include <hip/hip_runtime.h>
#include <math.h>

#define DD 128          // feature dim
#define HD 256          // heads * feature dim
#define HEADS 2
#define NEG_SLOPE 0.2f
#define BN_EPS 1e-5f

typedef float v2f __attribute__((ext_vector_type(2)));
typedef float v8f __attribute__((ext_vector_type(8)));

// ---------------------------------------------------------------------------
// GEMM: Hout[N, 256] = X[N, 128] @ W[128, 256] using V_WMMA_F32_16X16X4_F32.
// One wave per 16x16 output tile; 32 WMMA steps over K=128. All lanes active.
// ---------------------------------------------------------------------------
__global__ __launch_bounds__(128) void gat_gemm_wmma(
    const float* __restrict__ X, const float* __restrict__ W,
    float* __restrict__ Hout) {
  const int wave = threadIdx.x >> 5;
  const int lane = threadIdx.x & 31;
  const int tile = blockIdx.x * 4 + wave;   // total tiles = (N/16)*16 = N
  const int tm = tile >> 4;                 // row tile (16 col tiles for HD=256)
  const int tn = tile & 15;                 // col tile
  const int m0 = tm * 16;
  const int col = tn * 16 + (lane & 15);
  v8f c = {};
#if __has_builtin(__builtin_amdgcn_wmma_f32_16x16x4_f32)
  const int row = m0 + (lane & 15);
  const int kk  = (lane >> 4) * 2;          // lanes 16-31 hold K=2,3
  #pragma unroll 4
  for (int k = 0; k < DD; k += 4) {
    v2f a = *(const v2f*)(X + (size_t)row * DD + k + kk);  // A 16x4 frag
    v2f b;                                                 // B 4x16 frag
    b.x = W[(size_t)(k + kk)     * HD + col];
    b.y = W[(size_t)(k + kk + 1) * HD + col];
    c = __builtin_amdgcn_wmma_f32_16x16x4_f32(false, a, false, b,
                                              (short)0, c, false, false);
  }
  // C/D layout: VGPR r -> M = r (lanes 0-15) or M = r+8 (lanes 16-31), N = lane&15
  const int rbase = m0 + ((lane >> 4) << 3);
  #pragma unroll
  for (int r = 0; r < 8; ++r)
    Hout[(size_t)(rbase + r) * HD + col] = c[r];
#else
  const int rbase = m0 + ((lane >> 4) << 3);
  float acc[8] = {0.f, 0.f, 0.f, 0.f, 0.f, 0.f, 0.f, 0.f};
  for (int k = 0; k < DD; ++k) {
    float bv = W[(size_t)k * HD + col];
    #pragma unroll
    for (int r = 0; r < 8; ++r)
      acc[r] += X[(size_t)(rbase + r) * DD + k] * bv;
  }
  #pragma unroll
  for (int r = 0; r < 8; ++r)
    Hout[(size_t)(rbase + r) * HD + col] = acc[r];
#endif
}

// ---------------------------------------------------------------------------
// Per-node attention scores: a_s[n,h] = <h[n,h,:], att_src[h,:]>, same for dst.
// One wave (32 lanes x float4) per node.
// ---------------------------------------------------------------------------
__global__ __launch_bounds__(128) void gat_att_scores(
    const float* __restrict__ Hbuf, const float* __restrict__ att_s,
    const float* __restrict__ att_d, float* __restrict__ a_s,
    float* __restrict__ a_d, int n_nodes) {
  const int wave = threadIdx.x >> 5;
  const int lane = threadIdx.x & 31;
  const int n = blockIdx.x * 4 + wave;
  if (n >= n_nodes) return;
  #pragma unroll
  for (int h = 0; h < HEADS; ++h) {
    const float4 hv = *(const float4*)(Hbuf + (size_t)n * HD + h * DD + lane * 4);
    const float4 sv = *(const float4*)(att_s + h * DD + lane * 4);
    const float4 dv = *(const float4*)(att_d + h * DD + lane * 4);
    float ps = hv.x * sv.x + hv.y * sv.y + hv.z * sv.z + hv.w * sv.w;
    float pd = hv.x * dv.x + hv.y * dv.y + hv.z * dv.z + hv.w * dv.w;
    #pragma unroll
    for (int off = 16; off > 0; off >>= 1) {
      ps += __shfl_xor(ps, off, 32);
      pd += __shfl_xor(pd, off, 32);
    }
    if (lane == 0) {
      a_s[n * HEADS + h] = ps;
      a_d[n * HEADS + h] = pd;
    }
  }
}

// ---------------------------------------------------------------------------
// Init: out = bias (folds the "+bias" and empty-segment zero case), softmax
// state mmax = -inf, denom = 0.
// ---------------------------------------------------------------------------
__global__ void gat_init_layer(float* __restrict__ xout,
                               const float* __restrict__ bias,
                               float* __restrict__ mmax,
                               float* __restrict__ denom, int n_nodes) {
  size_t i = (size_t)blockIdx.x * blockDim.x + threadIdx.x;
  size_t tot = (size_t)n_nodes * DD;
  if (i < tot) xout[i] = bias[i & (DD - 1)];
  if (i < (size_t)n_nodes * HEADS) {
    mmax[i] = -__builtin_inff();
    denom[i] = 0.0f;
  }
}

__device__ __forceinline__ void atomicMaxFloat(float* addr, float val) {
  if (val >= 0.0f)
    atomicMax((int*)addr, __float_as_int(val));
  else
    atomicMin((unsigned int*)addr, __float_as_uint(val));
}

// ---------------------------------------------------------------------------
// Edge phase 1: raw leaky-relu scores -> exbuf, segment max into mmax[dst].
// ---------------------------------------------------------------------------
__global__ void gat_edge_scores_max(
    const int* __restrict__ src, const int* __restrict__ dst,
    const float* __restrict__ a_s, const float* __restrict__ a_d,
    float* __restrict__ exbuf, float* __restrict__ mmax, int n_edges) {
  int e = blockIdx.x * blockDim.x + threadIdx.x;
  if (e >= n_edges) return;
  int s = src[e], d = dst[e];
  float v0 = a_s[s * 2 + 0] + a_d[d * 2 + 0];
  float v1 = a_s[s * 2 + 1] + a_d[d * 2 + 1];
  v0 = v0 > 0.0f ? v0 : NEG_SLOPE * v0;
  v1 = v1 > 0.0f ? v1 : NEG_SLOPE * v1;
  *(float2*)(exbuf + 2 * (size_t)e) = make_float2(v0, v1);
  atomicMaxFloat(mmax + d * 2 + 0, v0);
  atomicMaxFloat(mmax + d * 2 + 1, v1);
}

// ---------------------------------------------------------------------------
// Edge phase 2: exbuf = exp(e - m[dst]); segment sum into denom[dst].
// ---------------------------------------------------------------------------
__global__ void gat_edge_exp_sum(
    const int* __restrict__ dst, float* __restrict__ exbuf,
    const float* __restrict__ mmax, float* __restrict__ denom, int n_edges) {
  int e = blockIdx.x * blockDim.x + threadIdx.x;
  if (e >= n_edges) return;
  int d = dst[e];
  float2 ev = *(float2*)(exbuf + 2 * (size_t)e);
  float x0 = __expf(ev.x - mmax[d * 2 + 0]);
  float x1 = __expf(ev.y - mmax[d * 2 + 1]);
  *(float2*)(exbuf + 2 * (size_t)e) = make_float2(x0, x1);
  atomicAdd(denom + d * 2 + 0, x0);
  atomicAdd(denom + d * 2 + 1, x1);
}

// ---------------------------------------------------------------------------
// Edge phase 3: message scatter. One block (128 threads) per edge; head-mean
// (x0.5) folded in so each (edge, d) issues exactly one f32 atomic into the
// L2-resident [N,128] output.
// ---------------------------------------------------------------------------
__global__ __launch_bounds__(128) void gat_scatter(
    const int* __restrict__ src, const int* __restrict__ dst,
    const float* __restrict__ Hbuf, const float* __restrict__ exbuf,
    const float* __restrict__ denom, float* __restrict__ xout) {
  const int e = blockIdx.x;
  const int t = threadIdx.x;
  const int s = src[e], d = dst[e];
  float2 ex = *(const float2*)(exbuf + 2 * (size_t)e);
  float2 dn = *(const float2*)(denom + 2 * (size_t)d);
  float al0 = ex.x / (dn.x + 1e-16f) * 0.5f;
  float al1 = ex.y / (dn.y + 1e-16f) * 0.5f;
  float v = al0 * Hbuf[(size_t)s * HD + t] + al1 * Hbuf[(size_t)s * HD + DD + t];
  atomicAdd(xout + (size_t)d * DD + t, v);
}

// ---------------------------------------------------------------------------
// BatchNorm (eval) + ReLU, in place.
// ---------------------------------------------------------------------------
__global__ void gat_bn_relu(float* __restrict__ x, const float* __restrict__ g,
                            const float* __restrict__ be,
                            const float* __restrict__ rm,
                            const float* __restrict__ rv, int n_nodes) {
  size_t i = (size_t)blockIdx.x * blockDim.x + threadIdx.x;
  if (i >= (size_t)n_nodes * DD) return;
  int d = (int)(i & (DD - 1));
  float v = (x[i] - rm[d]) * rsqrtf(rv[d] + BN_EPS) * g[d] + be[d];
  x[i] = v > 0.0f ? v : 0.0f;
}

// ---------------------------------------------------------------------------
// Host-side: one GAT layer = gemm -> att -> init -> max -> exp/sum -> scatter
// ---------------------------------------------------------------------------
static void run_gat_layer(const float* x_in, const float* Wl, const float* asl,
                          const float* adl, const float* bl, const int* src,
                          const int* dst, float* hbuf, float* a_s, float* a_d,
                          float* mmax, float* denom, float* exbuf, float* x_out,
                          int N, int E, hipStream_t stream) {
  // tiles = (N/16) row tiles * 16 col tiles = N; 4 waves (tiles) per block
  gat_gemm_wmma<<<N / 4, 128, 0, stream>>>(x_in, Wl, hbuf);
  gat_att_scores<<<(N + 3) / 4, 128, 0, stream>>>(hbuf, asl, adl, a_s, a_d, N);
  gat_init_layer<<<((size_t)N * DD + 255) / 256, 256, 0, stream>>>(x_out, bl,
                                                                   mmax, denom, N);
  gat_edge_scores_max<<<(E + 255) / 256, 256, 0, stream>>>(src, dst, a_s, a_d,
                                                           exbuf, mmax, E);
  gat_edge_exp_sum<<<(E + 255) / 256, 256, 0, stream>>>(dst, exbuf, mmax,
                                                        denom, E);
  gat_scatter<<<E, 128, 0, stream>>>(src, dst, hbuf, exbuf, denom, x_out);
}

extern "C" void kernel_launch(void* const* d_in, const int* in_sizes, int n_in,
                              void* d_out, int out_size, void* d_ws,
                              size_t ws_size, hipStream_t stream) {
  const int N = in_sizes[0] / DD;   // 50000
  const int E = in_sizes[1] / 2;    // 800000

  const float* x   = (const float*)d_in[0];
  const int* eidx  = (const int*)d_in[1];
  const int* src   = eidx;
  const int* dst   = eidx + E;
  const float* W[3]  = {(const float*)d_in[2], (const float*)d_in[6],
                        (const float*)d_in[10]};
  const float* AS[3] = {(const float*)d_in[3], (const float*)d_in[7],
                        (const float*)d_in[11]};
  const float* AD[3] = {(const float*)d_in[4], (const float*)d_in[8],
                        (const float*)d_in[12]};
  const float* B[3]  = {(const float*)d_in[5], (const float*)d_in[9],
                        (const float*)d_in[13]};
  const float* G[2]  = {(const float*)d_in[14], (const float*)d_in[18]};
  const float* BE[2] = {(const float*)d_in[15], (const float*)d_in[19]};
  const float* RM[2] = {(const float*)d_in[16], (const float*)d_in[20]};
  const float* RV[2] = {(const float*)d_in[17], (const float*)d_in[21]};

  // Workspace layout (floats): ~85 MB total
  float* ws    = (float*)d_ws;
  float* hbuf  = ws;                         // N*256
  float* xa    = hbuf + (size_t)N * HD;      // N*128 (ping buffer)
  float* a_s   = xa + (size_t)N * DD;        // N*2
  float* a_d   = a_s + (size_t)N * HEADS;    // N*2
  float* mmax  = a_d + (size_t)N * HEADS;    // N*2
  float* denom = mmax + (size_t)N * HEADS;   // N*2
  float* exbuf = denom + (size_t)N * HEADS;  // E*2

  // Layer 0: x -> xa, then BN+ReLU in place
  run_gat_layer(x, W[0], AS[0], AD[0], B[0], src, dst, hbuf, a_s, a_d, mmax,
                denom, exbuf, xa, N, E, stream);
  gat_bn_relu<<<((size_t)N * DD + 255) / 256, 256, 0, stream>>>(
      xa, G[0], BE[0], RM[0], RV[0], N);

  // Layer 1: xa -> xa (gemm consumes xa before init overwrites it; same stream)
  run_gat_layer(xa, W[1], AS[1], AD[1], B[1], src, dst, hbuf, a_s, a_d, mmax,
                denom, exbuf, xa, N, E, stream);
  gat_bn_relu<<<((size_t)N * DD + 255) / 256, 256, 0, stream>>>(
      xa, G[1], BE[1], RM[1], RV[1], N);

  // Layer 2: xa -> d_out (no BN/ReLU)
  run_gat_layer(xa, W[2], AS[2], AD[2], B[2], src, dst, hbuf, a_s, a_d, mmax,
                denom, exbuf, (float*)d_out, N, E, stream);
}